// LLaDARouter_21285857919730
// MI455X (gfx1250) — compile-verified
//
#include <hip/hip_runtime.h>
#include <math.h>

typedef __attribute__((ext_vector_type(2))) float v2f;
typedef __attribute__((ext_vector_type(8))) float v8f;
typedef unsigned int v4u __attribute__((ext_vector_type(4)));
typedef int v8i __attribute__((ext_vector_type(8)));
typedef int v4i __attribute__((ext_vector_type(4)));

#define D_DIM 2048
#define E_DIM 8
#define TOK_TILE 16
#define BLOCK 256
#define WAVES_PER_BLOCK (BLOCK / 32)

__device__ __forceinline__ float grp8_sum(float v) {
    v += __shfl_xor(v, 1, 32);
    v += __shfl_xor(v, 2, 32);
    v += __shfl_xor(v, 4, 32);
    return v;
}
__device__ __forceinline__ float grp8_max(float v) {
    v = fmaxf(v, __shfl_xor(v, 1, 32));
    v = fmaxf(v, __shfl_xor(v, 2, 32));
    v = fmaxf(v, __shfl_xor(v, 4, 32));
    return v;
}

__global__ void router_init_ws(float* ws) {
    if (threadIdx.x < 1 + E_DIM) ws[threadIdx.x] = 0.0f;
}

__global__ __launch_bounds__(BLOCK)
void router_main(const float* __restrict__ x, const float* __restrict__ W,
                 const float* __restrict__ gamma, const float* __restrict__ beta,
                 const float* __restrict__ temp,
                 float* __restrict__ out_rw, float* __restrict__ out_dp,
                 float* __restrict__ ws, int n_tok) {
    __shared__ __align__(16) float Wlds[E_DIM * D_DIM];   // 64 KB

    const int lane = threadIdx.x & 31;
    const int wave = threadIdx.x >> 5;

    // ---- Stage W (8 x 2048 fp32 = 64 KB) into LDS ----
#if __has_builtin(__builtin_amdgcn_tensor_load_to_lds)
    if (wave == 0) {
        // Tensor DMA descriptor: 1 "row" of 16384 fp32 elements, global -> LDS.
        const unsigned long long gaddr = (unsigned long long)(uintptr_t)W;
        const unsigned int lds_off = (unsigned int)(uintptr_t)&Wlds[0]; // low 32b = LDS addr
        v4u g0;
        g0[0] = 1u;                                        // count=1 (valid descriptor)
        g0[1] = lds_off;                                   // lds_addr, bits 63:32
        g0[2] = (unsigned int)(gaddr & 0xFFFFFFFFu);       // global_addr[31:0]
        g0[3] = (unsigned int)((gaddr >> 32) & 0x01FFFFFFu)// global_addr[56:32]
              | (2u << 30);                                // type=2 ("image")
        v8i g1 = {};
        g1[0] = 0x00020000;                   // data_size=2 (4B), no multicast/iterate/pad
        g1[1] = (int)(16384u << 16);          // tensor_dim0[15:0]=16384 in bits 63:48
        g1[2] = 0;                            // tensor_dim0[31:16]=0, tensor_dim1[15:0]=0
        g1[3] = (int)(16384u << 16);          // tile_dim0=16384 in bits 127:112
        g1[4] = 1;                            // tile_dim1=1, tile_dim2=0
        g1[5] = 16384;                        // tensor_dim0_stride[31:0]
        g1[6] = 0;                            // strides high bits
        g1[7] = 0;
        v4i gz = {};
#if __has_include(<hip/amd_detail/amd_gfx1250_TDM.h>)
        v8i gz8 = {};
        __builtin_amdgcn_tensor_load_to_lds(g0, g1, gz, gz, gz8, 0);   // 6-arg toolchain
#else
        __builtin_amdgcn_tensor_load_to_lds(g0, g1, gz, gz, 0);        // 5-arg toolchain
#endif
        __builtin_amdgcn_s_wait_tensorcnt(0);
    }
#else
    {
        const float4* src = (const float4*)W;
        float4* dst = (float4*)Wlds;
        for (int i = threadIdx.x; i < (E_DIM * D_DIM) / 4; i += BLOCK) dst[i] = src[i];
    }
#endif
    __syncthreads();

    const int e16  = lane & 15;          // expert column (8..15 duplicate 0..7)
    const int half = lane >> 4;          // 0: tokens r, 1: tokens r+8
    const int koff = half * 2;           // K offset inside the 4-wide step
    const bool valid = (e16 < E_DIM);

    const float g    = gamma[e16 & 7];
    const float bb   = beta[e16 & 7];
    const float invT = 1.0f / (fabsf(temp[0]) + 1e-6f);
    // Padded expert columns 8..15 read the same W row as columns 0..7: the
    // duplicate 8-lane groups then recompute identical (finite) results and
    // are simply never written -- no masking needed in the WMMA loop.
    const float* wrow = &Wlds[(e16 & 7) * D_DIM];

    const int n_tiles   = n_tok / TOK_TILE;
    const int gwave     = blockIdx.x * WAVES_PER_BLOCK + wave;
    const int tot_waves = gridDim.x * WAVES_PER_BLOCK;

    float zacc = 0.0f;   // sum of router_logits^2 (valid lanes only)
    float eacc = 0.0f;   // per-expert sum of routing weights

    for (int tile = gwave; tile < n_tiles; tile += tot_waves) {
        const int tbase = tile * TOK_TILE;
        const float* xrow = x + (size_t)(tbase + e16) * D_DIM + koff;

        v8f acc = {};
        #pragma unroll 8
        for (int kb = 0; kb < D_DIM; kb += 4) {
            v2f a = *(const v2f*)(xrow + kb);          // A: 16x4 (tokens x K)
            v2f b = *(const v2f*)(wrow + kb + koff);   // B: 4x16 (K x experts)
            acc = __builtin_amdgcn_wmma_f32_16x16x4_f32(
                false, a, false, b, (short)0, acc, false, false);
        }

        // Epilogue: C/D row r -> lanes 0-15 token tbase+r, lanes 16-31 tbase+r+8
        #pragma unroll
        for (int r = 0; r < 8; ++r) {
            float logit = acc[r];
            // LayerNorm over 8 experts (independent groups of 8 lanes)
            float mu  = grp8_sum(logit) * 0.125f;
            float dv  = logit - mu;
            float var = grp8_sum(dv * dv) * 0.125f;
            float ln  = dv * rsqrtf(var + 1e-5f) * g + bb;
            float rl  = ln * invT;
            // softmax over 8
            float mx = grp8_max(rl);
            float ex = __expf(rl - mx);
            float s  = grp8_sum(ex);
            float w  = ex / s;
            int e = e16 & 7;
            // top-1 (value-major, lower-index tie-break like jax top_k)
            float v1 = w; int i1 = e;
            #pragma unroll
            for (int m = 1; m <= 4; m <<= 1) {
                float ov = __shfl_xor(v1, m, 32);
                int   oi = __shfl_xor(i1, m, 32);
                if (ov > v1 || (ov == v1 && oi < i1)) { v1 = ov; i1 = oi; }
            }
            // top-2 (mask out winner; softmax weights > 0 so -1 acts as -inf)
            float v2 = (e == i1) ? -1.0f : w; int i2 = e;
            #pragma unroll
            for (int m = 1; m <= 4; m <<= 1) {
                float ov = __shfl_xor(v2, m, 32);
                int   oi = __shfl_xor(i2, m, 32);
                if (ov > v2 || (ov == v2 && oi < i2)) { v2 = ov; i2 = oi; }
            }
            float denom = v1 + v2 + 1e-6f;
            int tok = tbase + r + half * 8;
            if (valid) {
                out_rw[tok * E_DIM + e] = w;
                float dsp = (e == i1) ? (v1 / denom)
                          : (e == i2) ? (v2 / denom) : 0.0f;
                out_dp[tok * E_DIM + e] = dsp;
                zacc += rl * rl;
                eacc += w;
            }
        }
    }

    // z-loss partial: full-wave sum, one atomic per wave
    float zt = zacc;
    #pragma unroll
    for (int m = 1; m <= 16; m <<= 1) zt += __shfl_xor(zt, m, 32);
    if (lane == 0) atomicAdd(&ws[0], zt);
    // per-expert partials live in lanes e and e+16
    float et = eacc + __shfl_xor(eacc, 16, 32);
    if (lane < E_DIM) atomicAdd(&ws[1 + lane], et);
}

__global__ void router_finalize(const float* __restrict__ ws,
                                float* __restrict__ loss_out, int n_tok) {
    if (threadIdx.x == 0) {
        float z = ws[0] / (float)(n_tok * E_DIM);
        const float ideal = 1.0f / (float)E_DIM;
        float lb = 0.0f;
        for (int e = 0; e < E_DIM; ++e) {
            float actual = ws[1 + e] / (float)n_tok;
            lb += ideal * (logf(ideal) - logf(actual));
        }
        lb /= (float)E_DIM;
        loss_out[0] = 0.01f * z + 0.01f * lb;
    }
}

extern "C" void kernel_launch(void* const* d_in, const int* in_sizes, int n_in,
                              void* d_out, int out_size, void* d_ws, size_t ws_size,
                              hipStream_t stream) {
    const float* x     = (const float*)d_in[0];
    const float* W     = (const float*)d_in[1];
    const float* gamma = (const float*)d_in[2];
    const float* beta  = (const float*)d_in[3];
    const float* temp  = (const float*)d_in[4];

    const int n_tok = in_sizes[0] / D_DIM;          // 16384
    float* out    = (float*)d_out;
    float* out_rw = out;                            // [N, E]
    float* out_dp = out + (size_t)n_tok * E_DIM;    // [N, E]
    float* loss   = out + 2 * (size_t)n_tok * E_DIM;
    float* ws     = (float*)d_ws;                   // [0]=zsum, [1..8]=expert sums

    router_init_ws<<<1, 32, 0, stream>>>(ws);

    const int n_tiles = n_tok / TOK_TILE;                                  // 1024
    const int blocks  = (n_tiles + WAVES_PER_BLOCK - 1) / WAVES_PER_BLOCK; // 128
    router_main<<<blocks, BLOCK, 0, stream>>>(x, W, gamma, beta, temp,
                                              out_rw, out_dp, ws, n_tok);
    router_finalize<<<1, 32, 0, stream>>>(ws, loss, n_tok);
}